// PSO_7507602833740
// MI455X (gfx1250) — compile-verified
//
#include <hip/hip_runtime.h>

typedef __attribute__((ext_vector_type(2))) float v2f;
typedef __attribute__((ext_vector_type(8))) float v8f;

namespace {
constexpr int   kPairs = 28;
constexpr int   kIter  = 50;
constexpr float kScale = 16000.0f / 343.0f;   // samples per meter of path difference

__device__ __forceinline__ unsigned hash32(unsigned x) {
  x ^= x >> 16; x *= 0x7feb352dU;
  x ^= x >> 15; x *= 0x846ca68bU;
  x ^= x >> 16;
  return x;
}
__device__ __forceinline__ float u01(unsigned x) {
  return (float)(hash32(x) >> 8) * (1.0f / 16777216.0f);
}
} // namespace

// One wave32 per batch element; lane = particle (lanes 0..19 live).
// Loss:  loss_m = scale^2*(8*sum_r d^2 - (sum_r d)^2) - 2*scale*(d . w) + sum td^2
// d^2[m][r] computed with V_WMMA_F32_16X16X4_F32:
//   A[m][:] = [x,y,z,1],  B[:][r] = [-2rx,-2ry,-2rz,|r|^2+eps],  C[m][r] = |x_m|^2
__global__ __launch_bounds__(256) void pso_tdoa_kernel(
    const float* __restrict__ tdoas, const float* __restrict__ recv,
    float* __restrict__ out, int B) {
  __shared__ float stage[8 * 32];                 // one 32-slot row per wave
  const int lane  = threadIdx.x & 31;
  const int wslot = (threadIdx.x >> 5) * 32;
  const int b     = (int)blockIdx.x * 8 + (int)(threadIdx.x >> 5);
  if (b >= B) return;                             // wave-uniform

  stage[wslot + lane] = 0.0f;
  __builtin_prefetch(tdoas + (size_t)b * kPairs, 0, 1);   // global_prefetch_b8

  // centroid of the 8 receivers
  float cx = 0.f, cy = 0.f, cz = 0.f;
#pragma unroll
  for (int r = 0; r < 8; ++r) { cx += recv[3*r]; cy += recv[3*r+1]; cz += recv[3*r+2]; }
  cx *= 0.125f; cy *= 0.125f; cz *= 0.125f;

  const int  n  = lane & 15;          // D-layout column (receiver index)
  const bool hi = lane >= 16;

  // B operand (4x16 f32, 2 VGPRs): VGPR0 = K0 | K2, VGPR1 = K1 | K3
  float rnx = 0.f, rny = 0.f, rnz = 0.f, rn2 = 0.f;
  if (n < 8) {
    rnx = recv[3*n]; rny = recv[3*n+1]; rnz = recv[3*n+2];
    rn2 = rnx*rnx + rny*rny + rnz*rnz + 1e-12f;
  }
  v2f Bm;
  Bm.x = hi ? (-2.0f * rnz) : (-2.0f * rnx);      // K=2 : K=0
  Bm.y = hi ? rn2           : (-2.0f * rny);      // K=3 : K=1

  // Reduced TDOA terms: w_n and sum(td^2)
  constexpr int PI_[28] = {0,0,0,0,0,0,0,1,1,1,1,1,1,2,2,2,2,2,3,3,3,3,4,4,4,5,5,6};
  constexpr int PJ_[28] = {1,2,3,4,5,6,7,2,3,4,5,6,7,3,4,5,6,7,4,5,6,7,5,6,7,6,7,7};
  float wn = 0.0f, sumtd2 = 0.0f;
#pragma unroll
  for (int p = 0; p < kPairs; ++p) {
    float td = tdoas[(size_t)b * kPairs + p];
    sumtd2 += td * td;
    float sgn = (PI_[p] == n ? 1.0f : 0.0f) - (PJ_[p] == n ? 1.0f : 0.0f);
    wn += td * sgn;
  }
  if (n >= 8) wn = 0.0f;

  const float k8s2 = 8.0f * kScale * kScale;
  const float k2sw = 2.0f * kScale * wn;          // per-lane constant, hoisted
  const float s2c  = kScale * kScale;

  auto evalLoss = [&](float x0, float x1, float x2) -> float {
    const float s2   = x0*x0 + x1*x1 + x2*x2;
    const int   coff = hi ? 8 : 0;
    // ---- tile 0: particles 0..15 ----
    {
      int m = lane & 15;
      float sx0 = __shfl(x0, m), sx1 = __shfl(x1, m), sx2 = __shfl(x2, m);
      v2f A; A.x = hi ? sx2 : sx0; A.y = hi ? 1.0f : sx1;   // A 16x4 layout
      v8f C;
#pragma unroll
      for (int j = 0; j < 8; ++j) C[j] = __shfl(s2, j + coff);
      v8f D = __builtin_amdgcn_wmma_f32_16x16x4_f32(false, A, false, Bm,
                                                    (short)0, C, false, false);
#pragma unroll
      for (int j = 0; j < 8; ++j) {
        float dsq  = fmaxf(D[j], 0.0f);
        float dist = (n < 8) ? __builtin_amdgcn_sqrtf(dsq) : 0.0f;  // bare v_sqrt_f32
        float q    = (n < 8) ? (k8s2 * dsq - k2sw * dist) : 0.0f;
        q    += __shfl_xor(q, 1);    q    += __shfl_xor(q, 2);    q    += __shfl_xor(q, 4);
        dist += __shfl_xor(dist, 1); dist += __shfl_xor(dist, 2); dist += __shfl_xor(dist, 4);
        float lj = q - s2c * dist * dist + sumtd2;
        if (n == 0) stage[wslot + coff + j] = lj;   // particle coff+j
      }
    }
    // ---- tile 1: particles 16..19 (rows 0..3, low half of D) ----
    {
      int m = 16 + (lane & 15);
      float sx0 = __shfl(x0, m), sx1 = __shfl(x1, m), sx2 = __shfl(x2, m);
      v2f A; A.x = hi ? sx2 : sx0; A.y = hi ? 1.0f : sx1;
      v8f C;
#pragma unroll
      for (int j = 0; j < 8; ++j) C[j] = __shfl(s2, 16 + j + coff);
      v8f D = __builtin_amdgcn_wmma_f32_16x16x4_f32(false, A, false, Bm,
                                                    (short)0, C, false, false);
#pragma unroll
      for (int j = 0; j < 4; ++j) {
        float dsq  = fmaxf(D[j], 0.0f);
        float dist = (n < 8) ? __builtin_amdgcn_sqrtf(dsq) : 0.0f;
        float q    = (n < 8) ? (k8s2 * dsq - k2sw * dist) : 0.0f;
        q    += __shfl_xor(q, 1);    q    += __shfl_xor(q, 2);    q    += __shfl_xor(q, 4);
        dist += __shfl_xor(dist, 1); dist += __shfl_xor(dist, 2); dist += __shfl_xor(dist, 4);
        float lj = q - s2c * dist * dist + sumtd2;
        if (lane == 0) stage[wslot + 16 + j] = lj;  // particle 16+j
      }
    }
    asm volatile("s_wait_dscnt 0" ::: "memory");    // LDS RAW fence (split counter)
    return stage[wslot + lane];                     // lane-per-particle loss
  };

  // ---- PSO state ----
  float px = cx, py = cy, pz = cz;                  // x0 = centroid (all particles)
  unsigned seedbase = (unsigned)b * 0x9E3779B1u ^ ((unsigned)lane * 0x85EBCA77u);
  float vx = u01(seedbase ^ 0x1b873593u);
  float vy = u01(seedbase ^ 0xcc9e2d51u);
  float vz = u01(seedbase ^ 0xe6546b64u);
  float pbx = px, pby = py, pbz = pz;
  float l0  = evalLoss(px, py, pz);
  float pbl = (lane < 20) ? l0 : __builtin_inff();

  for (int it = 0; it < kIter; ++it) {
    // global best: argmin over lanes (lanes >= 20 hold +inf)
    float bv = pbl; int bi = lane;
#pragma unroll
    for (int m = 16; m >= 1; m >>= 1) {
      float ov = __shfl_xor(bv, m);
      int   oi = __shfl_xor(bi, m);
      if (ov < bv || (ov == bv && oi < bi)) { bv = ov; bi = oi; }
    }
    float gx = __shfl(pbx, bi), gy = __shfl(pby, bi), gz = __shfl(pbz, bi);

    unsigned s = seedbase + (unsigned)(it + 1) * 0x9E3779B9u;
    float r1x = u01(s ^ 0x0101u), r1y = u01(s ^ 0x0202u), r1z = u01(s ^ 0x0303u);
    float r2x = u01(s ^ 0x0404u), r2y = u01(s ^ 0x0505u), r2z = u01(s ^ 0x0606u);

    vx = 0.9f * vx + 0.5f * r1x * (pbx - px) + 0.3f * r2x * (gx - px);
    vy = 0.9f * vy + 0.5f * r1y * (pby - py) + 0.3f * r2y * (gy - py);
    vz = 0.9f * vz + 0.5f * r1z * (pbz - pz) + 0.3f * r2z * (gz - pz);
    px = fminf(fmaxf(px + vx, 0.1f), 10.0f);
    py = fminf(fmaxf(py + vy, 0.0f),  8.0f);
    pz = fminf(fmaxf(pz + vz, 1.0f),  2.0f);

    float l = evalLoss(px, py, pz);
    if (lane < 20 && l < pbl) { pbl = l; pbx = px; pby = py; pbz = pz; }
  }

  // final argmin + relative output
  float bv = pbl; int bi = lane;
#pragma unroll
  for (int m = 16; m >= 1; m >>= 1) {
    float ov = __shfl_xor(bv, m);
    int   oi = __shfl_xor(bi, m);
    if (ov < bv || (ov == bv && oi < bi)) { bv = ov; bi = oi; }
  }
  float gx = __shfl(pbx, bi), gy = __shfl(pby, bi), gz = __shfl(pbz, bi);
  if (lane == 0) {
    out[(size_t)b * 3 + 0] = gx - cx;
    out[(size_t)b * 3 + 1] = gy - cy;
    out[(size_t)b * 3 + 2] = gz - cz;
  }
}

extern "C" void kernel_launch(void* const* d_in, const int* in_sizes, int n_in,
                              void* d_out, int out_size, void* d_ws, size_t ws_size,
                              hipStream_t stream) {
  (void)n_in; (void)out_size; (void)d_ws; (void)ws_size;
  const float* tdoas = (const float*)d_in[0];   // (B, 28) f32
  const float* recv  = (const float*)d_in[1];   // (8, 3)  f32
  float* out = (float*)d_out;                   // (B, 3)  f32
  const int B = in_sizes[0] / kPairs;           // 8192
  const int blocks = (B + 7) / 8;               // 8 waves (8 batch elems) per block
  pso_tdoa_kernel<<<blocks, 256, 0, stream>>>(tdoas, recv, out, B);
}